// GraphAttentionLayer_4561255268882
// MI455X (gfx1250) — compile-verified
//
#include <hip/hip_runtime.h>
#include <hip/hip_bf16.h>

// GAT layer, fused flash-style, f32 WMMA (V_WMMA_F32_16X16X4_F32) throughout.
// Problem: ~9.1 GFLOP / ~35 MB traffic -> matrix-issue bound; fusion avoids the
// 128 MB attention materialization; adj (unused by reference) is never read.
//
// This revision stores all WMMA B-operands in a K-PAIRED LDS layout
// (pair[kp][col] = {X[2kp][col], X[2kp+1][col]}) so each lane's B-fragment is a
// single aligned ds_load_b64 into an even VGPR pair, eliminating the
// ds_load_2addr_b32 + v_dual_mov repacking seen in the previous build.

#define GAT_B 8
#define GAT_N 2048
#define GAT_C 128
#define ALPHA_SLOPE 0.2f

typedef __attribute__((ext_vector_type(2))) float v2f;
typedef __attribute__((ext_vector_type(8))) float v8f;

// ---------------------------------------------------------------------------
// Kernel 0: init per-batch s2max keys (order-preserving uint encoding of f32).
// ---------------------------------------------------------------------------
__global__ void gat_init_kernel(unsigned* __restrict__ s2maxkey) {
  if (threadIdx.x < GAT_B) s2maxkey[threadIdx.x] = 0u;
}

// ---------------------------------------------------------------------------
// Kernel 1: h = text @ W.   Grid (16,8) x 256 threads.
// W staged in LDS in K-paired form: Ws2[kp*128+col] = {W[2kp][col], W[2kp+1][col]}.
// Each wave: 16 rows x 128 cols (8 f-tiles), K-loop step 4 -> 256 wmma/wave.
// f32 WMMA operand mapping: lanes 0-15 take K={k,k+1}, lanes 16-31 K={k+2,k+3}
// -> kp = k/2 + (lane>=16).
// ---------------------------------------------------------------------------
__global__ __launch_bounds__(256) void gat_h_kernel(const float* __restrict__ text,
                                                    const float* __restrict__ W,
                                                    float* __restrict__ h) {
  __shared__ __align__(16) float Ws[64 * GAT_C * 2];  // 64 KB, float2-paired
  const int tid = threadIdx.x;

  // Pack W into K-paired LDS image (one-time; reads coalesced per row).
#pragma unroll
  for (int i = 0; i < 32; ++i) {
    const int idx = tid + 256 * i;  // float2 element index, 0..8191
    const int kp = idx >> 7;        // 0..63
    const int col = idx & 127;
    v2f v;
    v.x = W[(2 * kp) * GAT_C + col];
    v.y = W[(2 * kp + 1) * GAT_C + col];
    *(v2f*)&Ws[idx * 2] = v;  // ds_store_b64, coalesced
  }
  __syncthreads();

  const int wave = tid >> 5;
  const int lane = tid & 31;
  const int b = blockIdx.y;
  const int row = blockIdx.x * 128 + wave * 16 + (lane & 15);
  const int kpOff = lane >> 4;  // 0 for lanes 0-15, 1 for lanes 16-31
  const float* arow = text + ((size_t)b * GAT_N + row) * GAT_C;

  v8f acc[8];
#pragma unroll
  for (int ft = 0; ft < 8; ++ft) acc[ft] = {};

  for (int k = 0; k < GAT_C; k += 4) {
    v2f afrag = *(const v2f*)(arow + k + 2 * kpOff);  // global_load_b64
    const int kpBase = (k >> 1) + kpOff;
#pragma unroll
    for (int ft = 0; ft < 8; ++ft) {
      const int col = ft * 16 + (lane & 15);
      v2f bfrag = *(const v2f*)&Ws[(kpBase * GAT_C + col) * 2];  // ds_load_b64
      acc[ft] = __builtin_amdgcn_wmma_f32_16x16x4_f32(
          false, afrag, false, bfrag, (short)0, acc[ft], false, false);
    }
  }

  // C/D layout: lane holds col = lane&15, rows v + (lane>=16 ? 8 : 0)
  float* hbase = h + ((size_t)b * GAT_N + blockIdx.x * 128 + wave * 16) * GAT_C;
#pragma unroll
  for (int ft = 0; ft < 8; ++ft) {
    const int col = ft * 16 + (lane & 15);
#pragma unroll
    for (int v = 0; v < 8; ++v) {
      const int r = v + ((lane >> 4) << 3);
      hbase[r * GAT_C + col] = acc[ft][v];
    }
  }
}

// ---------------------------------------------------------------------------
// Kernel 2: s1[b,n]=h[b,n,:]·a1, s2[b,n]=h[b,n,:]·a2, s2max via uint atomicMax.
// ---------------------------------------------------------------------------
__global__ __launch_bounds__(256) void gat_score_kernel(const float* __restrict__ h,
                                                        const float* __restrict__ a,
                                                        float* __restrict__ s1,
                                                        float* __restrict__ s2,
                                                        unsigned* __restrict__ s2maxkey) {
  __shared__ float as[2 * GAT_C];
  const int tid = threadIdx.x;
  as[tid] = a[tid];
  __syncthreads();

  const int gid = blockIdx.x * 256 + tid;  // 0 .. B*N-1
  const float4* hrow = (const float4*)(h + (size_t)gid * GAT_C);
  float d1 = 0.f, d2 = 0.f;
#pragma unroll
  for (int q = 0; q < GAT_C / 4; ++q) {
    float4 hv = hrow[q];
    const int f = q * 4;
    d1 += hv.x * as[f] + hv.y * as[f + 1] + hv.z * as[f + 2] + hv.w * as[f + 3];
    d2 += hv.x * as[GAT_C + f] + hv.y * as[GAT_C + f + 1] +
          hv.z * as[GAT_C + f + 2] + hv.w * as[GAT_C + f + 3];
  }
  s1[gid] = d1;
  s2[gid] = d2;
  const unsigned bits = __float_as_uint(d2);
  const unsigned key = (bits & 0x80000000u) ? ~bits : (bits | 0x80000000u);
  atomicMax(&s2maxkey[gid >> 11], key);  // gid/2048 = batch
}

// ---------------------------------------------------------------------------
// Kernel 3: fused attention.  Grid (128,8) x 256 threads; WG = one 16-row tile.
// m_i = LR(s1_i + s2max) is the exact softmax max (LR strictly increasing):
// single pass accumulates num = sum_j w_j h_j (WMMA) and den = sum_j w_j
// (LDS float atomics).  h-tile staged K-paired; 512 wmma/wave.
// ---------------------------------------------------------------------------
#define JT 64
#define PS_STRIDE 66  // 64 + 2 pad (even -> aligned b64 A-frag loads)

__global__ __launch_bounds__(256) void gat_attn_kernel(const float* __restrict__ h,
                                                       const float* __restrict__ s1,
                                                       const float* __restrict__ s2,
                                                       const unsigned* __restrict__ s2maxkey,
                                                       float* __restrict__ out) {
  // K-paired h tile: hs2[jp*128+col] = {h[j0+2jp][col], h[j0+2jp+1][col]}
  __shared__ __align__(16) float hs[(JT / 2) * GAT_C * 2];  // 32 KB
  __shared__ __align__(16) float ps[16 * PS_STRIDE];        // ~4.2 KB
  __shared__ float den[16];

  const int tid = threadIdx.x;
  const int b = blockIdx.y;
  const int i0 = blockIdx.x * 16;

  const unsigned key = s2maxkey[b];
  const float s2max = (key & 0x80000000u) ? __uint_as_float(key & 0x7fffffffu)
                                          : __uint_as_float(~key);

  if (tid < 16) den[tid] = 0.f;

  const float* s1b = s1 + b * GAT_N;
  const float* s2b = s2 + b * GAT_N;
  const float* hb = h + (size_t)b * GAT_N * GAT_C;

  const int wave = tid >> 5;
  const int lane = tid & 31;
  const int kpOff = lane >> 4;
  const int colw = wave * 16 + (lane & 15);  // this wave's output column
  v8f acc = {};

  __syncthreads();  // den init visible

  for (int j0 = 0; j0 < GAT_N; j0 += JT) {
    // Stage h[j0:j0+64, :] K-paired: each thread interleaves 2 cols x 2 rows
    // (two global b64 loads -> one ds_store_b128).
#pragma unroll
    for (int rep = 0; rep < 8; ++rep) {
      const int idx = tid + 256 * rep;  // 0..2047
      const int jp = idx >> 6;          // 0..31
      const int c = (idx & 63) * 2;     // even col
      v2f lo = *(const v2f*)(hb + (size_t)(j0 + 2 * jp) * GAT_C + c);
      v2f hi = *(const v2f*)(hb + (size_t)(j0 + 2 * jp + 1) * GAT_C + c);
      float4 st = make_float4(lo.x, hi.x, lo.y, hi.y);
      *(float4*)&hs[(jp * GAT_C + c) * 2] = st;
    }
    // Build p tile: p[i][j] = exp(LR(s1_i+s2_j) - LR(s1_i+s2max)); den += p
#pragma unroll
    for (int rep = 0; rep < 4; ++rep) {
      const int idx = tid + 256 * rep;  // 0..1023
      const int i = idx >> 6;
      const int j = idx & 63;
      const float s1i = s1b[i0 + i];
      const float xm = s1i + s2max;
      const float mi = xm >= 0.f ? xm : ALPHA_SLOPE * xm;
      const float y = s1i + s2b[j0 + j];
      const float e = y >= 0.f ? y : ALPHA_SLOPE * y;
      const float p = __expf(e - mi);  // e - mi <= 0: no overflow
      ps[i * PS_STRIDE + j] = p;
      atomicAdd(&den[i], p);  // ds_add_f32
    }
    __syncthreads();
    // acc += P(16x64) @ H(64 x 16-col slab); both operands single b64 loads.
#pragma unroll
    for (int kk = 0; kk < JT; kk += 4) {
      v2f afrag = *(const v2f*)&ps[(lane & 15) * PS_STRIDE + kk + 2 * kpOff];
      v2f bfrag = *(const v2f*)&hs[(((kk >> 1) + kpOff) * GAT_C + colw) * 2];
      acc = __builtin_amdgcn_wmma_f32_16x16x4_f32(
          false, afrag, false, bfrag, (short)0, acc, false, false);
    }
    __syncthreads();  // protects hs/ps reuse; after last iter, den is final
  }

  // h' = acc/den ; out = elu(h' + 0.2*h)
#pragma unroll
  for (int v = 0; v < 8; ++v) {
    const int r = v + ((lane >> 4) << 3);
    const float hp = acc[v] / den[r];
    const float hv = hb[(size_t)(i0 + r) * GAT_C + colw];
    const float z = hp + ALPHA_SLOPE * hv;
    out[((size_t)b * GAT_N + i0 + r) * GAT_C + colw] =
        z > 0.f ? z : (__expf(z) - 1.f);
  }
}

// ---------------------------------------------------------------------------
extern "C" void kernel_launch(void* const* d_in, const int* in_sizes, int n_in,
                              void* d_out, int out_size, void* d_ws, size_t ws_size,
                              hipStream_t stream) {
  (void)in_sizes; (void)n_in; (void)out_size; (void)ws_size;
  const float* text = (const float*)d_in[0];
  // d_in[1] = adj: unused by the reference computation
  const float* W = (const float*)d_in[2];
  const float* a = (const float*)d_in[3];
  float* out = (float*)d_out;

  char* ws = (char*)d_ws;
  float* h = (float*)ws;                                         // 8 MB
  float* s1 = (float*)(ws + (size_t)GAT_B * GAT_N * GAT_C * 4);  // 64 KB
  float* s2 = s1 + GAT_B * GAT_N;                                // 64 KB
  unsigned* s2maxkey = (unsigned*)(s2 + GAT_B * GAT_N);          // 32 B

  gat_init_kernel<<<dim3(1), dim3(32), 0, stream>>>(s2maxkey);
  gat_h_kernel<<<dim3(GAT_N / 128, GAT_B), dim3(256), 0, stream>>>(text, W, h);
  gat_score_kernel<<<dim3(GAT_B * GAT_N / 256), dim3(256), 0, stream>>>(h, a, s1, s2, s2maxkey);
  gat_attn_kernel<<<dim3(GAT_N / 16, GAT_B), dim3(256), 0, stream>>>(h, s1, s2, s2maxkey, out);
}